// GerDALoss_32615981646487
// MI455X (gfx1250) — compile-verified
//
#include <hip/hip_runtime.h>

typedef __attribute__((ext_vector_type(2))) float v2f;
typedef __attribute__((ext_vector_type(8))) float v8f;

static constexpr int CLS = 10;   // classes
static constexpr int DIM = 64;   // feature dim

// workspace layout (floats): [0,4096) ZtZ ; [4096,4736) class sums ; [4736,4746) counts
static constexpr int WS_ZTZ = 0;
static constexpr int WS_SC  = DIM * DIM;
static constexpr int WS_NC  = DIM * DIM + CLS * DIM;
static constexpr int WS_TOT = DIM * DIM + CLS * DIM + CLS;

__global__ void gerda_zero(float* __restrict__ ws, int n) {
  for (int i = blockIdx.x * blockDim.x + threadIdx.x; i < n;
       i += gridDim.x * blockDim.x)
    ws[i] = 0.0f;
}

// Pass 1: one streaming pass over Z. Per wave: syrk of 4 rows via 16x f32
// WMMA (acc += Zc^T Zc). Class sums are accumulated directly from the WMMA
// fragment registers into LDS (no second read of Z). Flush via atomics.
__global__ __launch_bounds__(256) void gerda_pass1(const float* __restrict__ Z,
                                                   const int* __restrict__ Y,
                                                   float* __restrict__ ws,
                                                   int nChunks) {
  __shared__ float sAcc[DIM * DIM];   // 16 KB block-level ZtZ reduction
  __shared__ float sSc[CLS * DIM];    // class sums
  __shared__ float sNc[CLS];          // class counts (float)

  const int tid = threadIdx.x;
  for (int i = tid; i < CLS * DIM; i += 256) sSc[i] = 0.0f;
  if (tid < CLS) sNc[tid] = 0.0f;
  for (int i = tid; i < DIM * DIM; i += 256) sAcc[i] = 0.0f;
  __syncthreads();

  const int lane = tid & 31;
  const int wv   = tid >> 5;          // 8 waves per block
  const int dl   = lane & 15;         // M index within A fragment
  const int kb   = (lane >> 4) << 1;  // K base: 0 (lanes 0-15) or 2 (lanes 16-31)

  v8f acc[16] = {};                   // 4x4 tiles of 16x16 f32 accumulators

  for (int chunk = blockIdx.x; chunk < nChunks; chunk += gridDim.x) {
    const long base = (long)chunk * 32;
    const long rA   = base + (long)wv * 4 + kb;   // this lane's first row

    // ---- fragment loads: lane holds cols {dl,dl+16,dl+32,dl+48} of rows rA, rA+1
    const float* p = Z + rA * DIM + dl;
    v2f frag[4];
#pragma unroll
    for (int m = 0; m < 4; ++m) {
      frag[m].x = p[16 * m];          // row rA
      frag[m].y = p[DIM + 16 * m];    // row rA+1
    }
    const int y0 = Y[rA];             // one coalesced dword per wave
    const int y1 = Y[rA + 1];

    // ---- WMMA syrk: acc(m,n) += Zc^T(m-tile) * Zc(n-tile) ----
#pragma unroll
    for (int m = 0; m < 4; ++m)
#pragma unroll
      for (int n = 0; n < 4; ++n)
        acc[m * 4 + n] = __builtin_amdgcn_wmma_f32_16x16x4_f32(
            false, frag[m], false, frag[n], (short)0, acc[m * 4 + n],
            false, false);

    // ---- class sums straight from the fragment registers ----
#pragma unroll
    for (int m = 0; m < 4; ++m) {
      atomicAdd(&sSc[y0 * DIM + 16 * m + dl], frag[m].x);
      atomicAdd(&sSc[y1 * DIM + 16 * m + dl], frag[m].y);
    }
    if (dl == 0) {                    // each of the 4 rows counted once per wave
      atomicAdd(&sNc[y0], 1.0f);
      atomicAdd(&sNc[y1], 1.0f);
    }
  }
  __syncthreads();

  // wave-private accumulators -> block LDS (C/D layout: row = m*16 + v + 8*(lane>=16))
  const int rb = (lane >> 4) * 8;
  const int c0 = lane & 15;
#pragma unroll
  for (int m = 0; m < 4; ++m)
#pragma unroll
    for (int n = 0; n < 4; ++n)
#pragma unroll
      for (int v = 0; v < 8; ++v)
        atomicAdd(&sAcc[(m * 16 + rb + v) * DIM + n * 16 + c0],
                  acc[m * 4 + n][v]);
  __syncthreads();

  for (int i = tid; i < DIM * DIM; i += 256) atomicAdd(&ws[WS_ZTZ + i], sAcc[i]);
  for (int i = tid; i < CLS * DIM; i += 256) atomicAdd(&ws[WS_SC + i], sSc[i]);
  if (tid < CLS) atomicAdd(&ws[WS_NC + tid], sNc[tid]);
}

// Pass 2: single block. Tail rows, means, Sb, St, 64x64 Gauss-Jordan inverse,
// trace contraction -> scalar.
__global__ __launch_bounds__(256) void gerda_pass2(const float* __restrict__ Z,
                                                   const int* __restrict__ Y,
                                                   const float* __restrict__ ws,
                                                   float* __restrict__ out,
                                                   int N, int tailStart) {
  __shared__ float G[DIM * 128];      // [St | I] -> [I | inv(St)]   32 KB
  __shared__ float sSb[DIM * DIM];    // 16 KB
  __shared__ float sMc[CLS * DIM];    // class sums -> means
  __shared__ float sNc[CLS];
  __shared__ float sTr[CLS * CLS];
  __shared__ float sW[CLS * CLS];
  __shared__ float szrow[DIM];
  __shared__ float sRed[256];
  __shared__ float sPiv;

  const int tid = threadIdx.x;
  const float Nf = (float)N;

  for (int i = tid; i < DIM * DIM; i += 256)
    G[(i >> 6) * 128 + (i & 63)] = ws[WS_ZTZ + i];
  for (int i = tid; i < CLS * DIM; i += 256) sMc[i] = ws[WS_SC + i];
  if (tid < CLS) sNc[tid] = ws[WS_NC + tid];
  __syncthreads();

  // scalar tail (rows not covered by 32-row chunks)
  for (int row = tailStart; row < N; ++row) {
    if (tid < DIM) szrow[tid] = Z[(long)row * DIM + tid];
    __syncthreads();
    const int y = Y[row];
    if (tid < DIM) sMc[y * DIM + tid] += szrow[tid];
    if (tid == 0) sNc[y] += 1.0f;
    for (int i = tid; i < DIM * DIM; i += 256)
      G[(i >> 6) * 128 + (i & 63)] += szrow[i >> 6] * szrow[i & 63];
    __syncthreads();
  }

  // class means
  for (int i = tid; i < CLS * DIM; i += 256) sMc[i] = sMc[i] / sNc[i >> 6];
  __syncthreads();

  // pairwise squared distances between means
  if (tid < CLS * CLS) {
    const int ci = tid / CLS, cj = tid % CLS;
    float t = 0.0f;
    for (int d = 0; d < DIM; ++d) {
      const float df = sMc[ci * DIM + d] - sMc[cj * DIM + d];
      t += df * df;
    }
    sTr[tid] = t;
  }
  __syncthreads();

  // Db (weighted), Db.max, fold all scalars into pair weights
  if (tid == 0) {
    float dbmax = 0.0f;
    for (int k = 0; k < CLS * CLS; ++k) {
      const int ci = k / CLS, cj = k % CLS;
      float db = 0.0f;
      if (ci != cj) {
        const float t = sTr[k];
        db = 1.0f / (t > 0.0f ? t : 1.0f);
      }
      sW[k] = db;
      dbmax = fmaxf(dbmax, db);
    }
    const float scale = 0.5f / (Nf * Nf * dbmax);
    for (int k = 0; k < CLS * CLS; ++k) {
      const int ci = k / CLS, cj = k % CLS;
      sW[k] = sNc[ci] * sNc[cj] * sW[k] * scale;
    }
  }
  __syncthreads();

  // Sb[d,e] = sum_ij w_ij * diff_ij[d] * diff_ij[e]
  for (int i = tid; i < DIM * DIM; i += 256) {
    const int d = i >> 6, e = i & 63;
    float s = 0.0f;
    for (int k = 0; k < CLS * CLS; ++k) {
      const int ci = k / CLS, cj = k % CLS;
      s += sW[k] * (sMc[ci * DIM + d] - sMc[cj * DIM + d]) *
                   (sMc[ci * DIM + e] - sMc[cj * DIM + e]);
    }
    sSb[i] = s;
  }
  __syncthreads();

  // St = Sw + Sb  (Sw = (ZtZ - sum_c Nc mc mc^T)/N), identity in right half
  for (int i = tid; i < DIM * DIM; i += 256) {
    const int d = i >> 6, e = i & 63;
    float cc = 0.0f;
    for (int c = 0; c < CLS; ++c)
      cc += sNc[c] * sMc[c * DIM + d] * sMc[c * DIM + e];
    const float sw = (G[d * 128 + e] - cc) / Nf;
    G[d * 128 + e] = sw + sSb[i];
    G[d * 128 + 64 + e] = (d == e) ? 1.0f : 0.0f;
  }
  __syncthreads();

  // Gauss-Jordan inversion (St is SPD -> no pivoting)
  const int r   = tid >> 2;           // 4 threads per row
  const int c0e = (tid & 3) * 32;     // each covers 32 of 128 cols
  for (int k = 0; k < DIM; ++k) {
    if (tid == 0) sPiv = 1.0f / G[k * 128 + k];
    __syncthreads();
    if (tid < 128) G[k * 128 + tid] *= sPiv;
    __syncthreads();
    const float f = G[r * 128 + k];
    __syncthreads();
    if (r != k)
      for (int j = c0e; j < c0e + 32; ++j)
        G[r * 128 + j] -= f * G[k * 128 + j];
    __syncthreads();
  }

  // trace(inv(St) @ Sb) = sum_{d,e} invSt[d,e] * Sb[e,d]
  float part = 0.0f;
  for (int i = tid; i < DIM * DIM; i += 256) {
    const int d = i >> 6, e = i & 63;
    part += G[d * 128 + 64 + e] * sSb[e * DIM + d];
  }
  sRed[tid] = part;
  __syncthreads();
  for (int s = 128; s > 0; s >>= 1) {
    if (tid < s) sRed[tid] += sRed[tid + s];
    __syncthreads();
  }
  if (tid == 0) out[0] = 1.0f - sRed[0] / (float)DIM;
}

extern "C" void kernel_launch(void* const* d_in, const int* in_sizes, int n_in,
                              void* d_out, int out_size, void* d_ws, size_t ws_size,
                              hipStream_t stream) {
  const float* Z = (const float*)d_in[0];
  const int*   Y = (const int*)d_in[1];
  float* out = (float*)d_out;
  float* ws  = (float*)d_ws;

  const int N = in_sizes[1];          // Y element count
  const int nChunks = N / 32;

  gerda_zero<<<8, 256, 0, stream>>>(ws, WS_TOT);
  if (nChunks > 0) {
    const int grid = nChunks < 512 ? nChunks : 512;
    gerda_pass1<<<grid, 256, 0, stream>>>(Z, Y, ws, nChunks);
  }
  gerda_pass2<<<1, 256, 0, stream>>>(Z, Y, ws, out, N, nChunks * 32);
}